// VIF_3109556322610
// MI455X (gfx1250) — compile-verified
//
#include <hip/hip_runtime.h>
#include <hip/hip_bf16.h>

typedef __attribute__((ext_vector_type(2))) float v2f;
typedef __attribute__((ext_vector_type(8))) float v8f;

__device__ __forceinline__ v8f wmma4(v2f a, v2f b, v8f c) {
    // D(16x16,f32) += A(16x4,f32) * B(4x16,f32)
    return __builtin_amdgcn_wmma_f32_16x16x4_f32(
        /*neg_a=*/false, a, /*neg_b=*/false, b,
        /*c_mod=*/(short)0, c, /*reuse_a=*/false, /*reuse_b=*/false);
}

__device__ __forceinline__ int reflect_idx(int i, int n) {
    if (i < 0) i = -i;
    if (i >= n) i = 2 * n - 2 - i;
    return i;
}

__device__ __forceinline__ float chval(float r, float d, int ch) {
    if (ch == 0) return r;
    if (ch == 1) return d;
    if (ch == 2) return r * r;
    if (ch == 3) return d * d;
    return r * d;
}

__device__ __forceinline__ void make_gauss_lds(float* g1d, int N, int lane) {
    if (lane == 0) {
        float sigma = (float)N / 5.0f;
        float inv2s2 = 1.0f / (2.0f * sigma * sigma);
        float s = 0.0f;
        for (int k = 0; k < N; ++k) {
            float x = (float)k - (float)(N - 1) * 0.5f;
            float e = __expf(-(x * x) * inv2s2);
            g1d[k] = e;
            s += e;
        }
        float inv = 1.0f / s;
        for (int k = 0; k < N; ++k) g1d[k] *= inv;
    }
}

// ---------------------------------------------------------------------------
// Stats kernel: one wave (32 threads) per 16x16 output tile per batch image.
// Separable grouped conv of 5 channels as banded-GEMMs on V_WMMA_F32_16X16X4_F32.
// STEPS = ceil((16+2p)/4) K-chunks; band weights live in a zero-padded LDS
// table so all fragment builds are branch-free unconditional LDS loads, and
// the same per-lane weight fragment serves both the horizontal (Bh[j][c]=g[j-c])
// and vertical (Wv[r][j]=g[j-r]) passes.
// ---------------------------------------------------------------------------
template <int STEPS>
__global__ __launch_bounds__(32) void vif_stats_kernel(
    const float* __restrict__ ref, const float* __restrict__ dist,
    int h, int w, int N, int p, int sub128, int scale,
    float* __restrict__ acc) {

    __shared__ float gpad[64];       // zero-padded; weights at gpad[16+k]
    __shared__ float tR[32][33];
    __shared__ float tD[32][33];
    __shared__ float Hb[32][17];

    const int lane = threadIdx.x;

    // zero-fill weight table and halo tiles (padding regions must be 0)
    for (int i = lane; i < 64; i += 32) gpad[i] = 0.0f;
    for (int i = lane; i < 32 * 33; i += 32) {
        (&tR[0][0])[i] = 0.0f;
        (&tD[0][0])[i] = 0.0f;
    }
    __syncthreads();

    make_gauss_lds(gpad + 16, N, lane);   // lane 0 writes N taps

    const int KT = 16 + 2 * p;            // halo tile side, <= 32
    const int r0 = blockIdx.y * 16;
    const int c0 = blockIdx.x * 16;
    const float* R = ref  + (long long)blockIdx.z * h * w;
    const float* D = dist + (long long)blockIdx.z * h * w;

    for (int i2 = lane; i2 < KT * KT; i2 += 32) {
        int rr = i2 / KT, cc = i2 - rr * KT;
        int gr = reflect_idx(r0 - p + rr, h);
        int gc = reflect_idx(c0 - p + cc, w);
        float vr = R[(long long)gr * w + gc];
        float vd = D[(long long)gr * w + gc];
        if (sub128) { vr -= 128.0f; vd -= 128.0f; }
        tR[rr][cc] = vr;
        tD[rr][cc] = vd;
    }
    __syncthreads();

    const int mrow = lane & 15;
    const int kofs = (lane < 16) ? 0 : 2;   // A/B K split across lane halves
    const int rofs = (lane < 16) ? 0 : 8;   // C/D row split across lane halves

    // Per-lane banded weight fragments, shared by both conv passes.
    v2f wband[STEPS];
#pragma unroll
    for (int k = 0; k < STEPS; ++k) {
        int kA = 4 * k + kofs;
        wband[k].x = gpad[16 + kA - mrow];
        wband[k].y = gpad[16 + kA + 1 - mrow];
    }

    v8f cacc[5];

#pragma unroll
    for (int ch = 0; ch < 5; ++ch) {
        __syncthreads();   // protect Hb against previous channel's readers

        // ---- horizontal: H[32 x 16] = ChTile[32 x 4S] * Bh[4S x 16] ----
        v8f h0 = {};   // H rows 0..15
        v8f h1 = {};   // H rows 16..31
#pragma unroll
        for (int k = 0; k < STEPS; ++k) {
            int kA = 4 * k + kofs;
            v2f a0, a1;
            a0.x = chval(tR[mrow][kA],          tD[mrow][kA],          ch);
            a0.y = chval(tR[mrow][kA + 1],      tD[mrow][kA + 1],      ch);
            a1.x = chval(tR[mrow + 16][kA],     tD[mrow + 16][kA],     ch);
            a1.y = chval(tR[mrow + 16][kA + 1], tD[mrow + 16][kA + 1], ch);
            h0 = wmma4(a0, wband[k], h0);
            h1 = wmma4(a1, wband[k], h1);
        }
#pragma unroll
        for (int i = 0; i < 8; ++i) {
            Hb[i + rofs][mrow]      = h0[i];
            Hb[16 + i + rofs][mrow] = h1[i];
        }
        __syncthreads();

        // ---- vertical: Out[16 x 16] = Wv[16 x 4S] * H[4S x 16] ----
        v8f vac = {};
#pragma unroll
        for (int k = 0; k < STEPS; ++k) {
            int kA = 4 * k + kofs;
            v2f bb;
            bb.x = Hb[kA][mrow];
            bb.y = Hb[kA + 1][mrow];
            vac = wmma4(wband[k], bb, vac);
        }
        cacc[ch] = vac;
    }

    // ---- per-pixel VIF statistics (exact reference masking order) ----
    float numSum = 0.0f, denSum = 0.0f;
#pragma unroll
    for (int i = 0; i < 8; ++i) {
        float mu1 = cacc[0][i], mu2 = cacc[1][i];
        float s1  = fmaxf(cacc[2][i] - mu1 * mu1, 0.0f);
        float s2  = fmaxf(cacc[3][i] - mu2 * mu2, 0.0f);
        float s12 = cacc[4][i] - mu1 * mu2;
        float gg  = s12 / (s1 + 1e-10f);
        float sv  = s2 - gg * s12;
        if (s1 < 1e-10f) { gg = 0.0f; sv = s2; s1 = 0.0f; }
        if (s2 < 1e-10f) { gg = 0.0f; sv = 0.0f; }
        if (gg < 0.0f) sv = s2;
        gg = fmaxf(gg, 0.0f);
        if (sv <= 1e-10f) sv = 1e-10f;
        gg = fminf(gg, 100.0f);
        float num = log2f(1.0f + gg * gg * s1 / (sv + 2.0f));
        float den = log2f(1.0f + s1 * 0.5f);
        if (s12 < 0.0f) num = 0.0f;
        if (s1 < 2.0f) { num = 1.0f - s2 * (4.0f / (255.0f * 255.0f)); den = 1.0f; }
        int gr = r0 + i + rofs;
        int gc = c0 + mrow;
        if (gr < h && gc < w) { numSum += num; denSum += den; }
    }

    // wave32 reduction
    for (int off = 16; off > 0; off >>= 1) {
        numSum += __shfl_down(numSum, off, 32);
        denSum += __shfl_down(denSum, off, 32);
    }
    if (lane == 0) {
        int ai = blockIdx.z * 4 + scale;
        atomicAdd(&acc[ai], numSum);
        atomicAdd(&acc[16 + ai], denSum);
    }
}

// ---------------------------------------------------------------------------
// Stride-2 reflect-padded Gaussian downsample (tiny FLOPs -> plain VALU).
// ---------------------------------------------------------------------------
__global__ __launch_bounds__(256) void vif_down_kernel(
    const float* __restrict__ src, float* __restrict__ dst,
    int hin, int win_, int hout, int wout, int N, int p, int sub128, int batch) {

    __shared__ float g1d[24];
    make_gauss_lds(g1d, N, threadIdx.x);
    __syncthreads();

    long long total = (long long)batch * hout * wout;
    long long idx = (long long)blockIdx.x * blockDim.x + threadIdx.x;
    if (idx >= total) return;
    int b   = (int)(idx / ((long long)hout * wout));
    int rem = (int)(idx - (long long)b * hout * wout);
    int y = rem / wout, x = rem - y * wout;

    const float* sp = src + (long long)b * hin * win_;
    float accv = 0.0f;
    for (int i = 0; i < N; ++i) {
        int ry = reflect_idx(2 * y + i - p, hin);
        float row = 0.0f;
        for (int j = 0; j < N; ++j) {
            int rx = reflect_idx(2 * x + j - p, win_);
            float v = sp[(long long)ry * win_ + rx];
            if (sub128) v -= 128.0f;
            row += g1d[j] * v;
        }
        accv += g1d[i] * row;
    }
    dst[idx] = accv;
}

__global__ void vif_init_kernel(float* __restrict__ acc) {
    if (threadIdx.x < 32) acc[threadIdx.x] = 0.0f;
}

__global__ void vif_final_kernel(const float* __restrict__ acc, float* __restrict__ out) {
    int i = threadIdx.x;
    if (i < 16) out[i] = acc[i] / acc[16 + i];
}

// ---------------------------------------------------------------------------
extern "C" void kernel_launch(void* const* d_in, const int* in_sizes, int n_in,
                              void* d_out, int out_size, void* d_ws, size_t ws_size,
                              hipStream_t stream) {
    const float* ref  = (const float*)d_in[0];
    const float* dist = (const float*)d_in[1];
    float* out = (float*)d_out;
    float* ws  = (float*)d_ws;

    const int H0 = 1080, W0 = 1920;
    const int B = in_sizes[0] / (H0 * W0);
    const int H1 = H0 / 2, W1 = W0 / 2;   // 540 x 960
    const int H2 = H1 / 2, W2 = W1 / 2;   // 270 x 480
    const int H3 = H2 / 2, W3 = W2 / 2;   // 135 x 240

    // workspace layout (floats)
    float* acc = ws;                                  // [0..15]=num, [16..31]=den
    size_t off = 32;
    float* ref1  = ws + off; off += (size_t)B * H1 * W1;
    float* dist1 = ws + off; off += (size_t)B * H1 * W1;
    float* ref2  = ws + off; off += (size_t)B * H2 * W2;
    float* dist2 = ws + off; off += (size_t)B * H2 * W2;
    float* ref3  = ws + off; off += (size_t)B * H3 * W3;
    float* dist3 = ws + off; off += (size_t)B * H3 * W3;
    (void)ws_size; (void)n_in; (void)out_size;

    auto tiles = [](int n) { return (n + 15) / 16; };
    auto dblk  = [](long long t) { return (unsigned)((t + 255) / 256); };

    vif_init_kernel<<<1, 32, 0, stream>>>(acc);

    // scale 0: N=17, p=8 -> KT=32 -> 8 K-steps; raw input (subtract 128)
    vif_stats_kernel<8><<<dim3(tiles(W0), tiles(H0), B), 32, 0, stream>>>(
        ref, dist, H0, W0, 17, 8, 1, 0, acc);

    // scale 1: downsample with N=9 window, then stats (KT=24 -> 6 K-steps)
    vif_down_kernel<<<dblk((long long)B * H1 * W1), 256, 0, stream>>>(
        ref,  ref1,  H0, W0, H1, W1, 9, 4, 1, B);
    vif_down_kernel<<<dblk((long long)B * H1 * W1), 256, 0, stream>>>(
        dist, dist1, H0, W0, H1, W1, 9, 4, 1, B);
    vif_stats_kernel<6><<<dim3(tiles(W1), tiles(H1), B), 32, 0, stream>>>(
        ref1, dist1, H1, W1, 9, 4, 0, 1, acc);

    // scale 2: N=5 (KT=20 -> 5 K-steps)
    vif_down_kernel<<<dblk((long long)B * H2 * W2), 256, 0, stream>>>(
        ref1,  ref2,  H1, W1, H2, W2, 5, 2, 0, B);
    vif_down_kernel<<<dblk((long long)B * H2 * W2), 256, 0, stream>>>(
        dist1, dist2, H1, W1, H2, W2, 5, 2, 0, B);
    vif_stats_kernel<5><<<dim3(tiles(W2), tiles(H2), B), 32, 0, stream>>>(
        ref2, dist2, H2, W2, 5, 2, 0, 2, acc);

    // scale 3: N=3 (KT=18 -> 5 K-steps, padded; padding is exact zeros)
    vif_down_kernel<<<dblk((long long)B * H3 * W3), 256, 0, stream>>>(
        ref2,  ref3,  H2, W2, H3, W3, 3, 1, 0, B);
    vif_down_kernel<<<dblk((long long)B * H3 * W3), 256, 0, stream>>>(
        dist2, dist3, H2, W2, H3, W3, 3, 1, 0, B);
    vif_stats_kernel<5><<<dim3(tiles(W3), tiles(H3), B), 32, 0, stream>>>(
        ref3, dist3, H3, W3, 3, 1, 0, 3, acc);

    vif_final_kernel<<<1, 16, 0, stream>>>(acc, out);
}